// GraphPooling_6906307412352
// MI455X (gfx1250) — compile-verified
//
#include <hip/hip_runtime.h>
#include <stdint.h>

#define POOLK 4
#define GDIM  64
#define G3    (GDIM*GDIM*GDIM)   // 262144

typedef unsigned int u32;

// ---------- order-preserving float <-> uint bijection (for atomic max) ----------
__device__ __forceinline__ u32 f2ord(float f) {
    u32 u = __float_as_uint(f);
    return (u & 0x80000000u) ? ~u : (u | 0x80000000u);
}
__device__ __forceinline__ float ord2f(u32 u) {
    u = (u & 0x80000000u) ? (u & 0x7FFFFFFFu) : ~u;
    return __uint_as_float(u);
}

// ---------- utility fills ----------
__global__ void k_zero_u32(u32* p, long long n) {
    long long i = (long long)blockIdx.x * blockDim.x + threadIdx.x;
    long long st = (long long)gridDim.x * blockDim.x;
    for (; i < n; i += st) p[i] = 0u;
}
__global__ void k_fill_f32(float* p, float v, long long n) {
    long long i = (long long)blockIdx.x * blockDim.x + threadIdx.x;
    long long st = (long long)gridDim.x * blockDim.x;
    for (; i < n; i += st) p[i] = v;
}
__global__ void k_copy_u32(u32* dst, const u32* src, int n) {
    int i = blockIdx.x * blockDim.x + threadIdx.x;
    if (i < n) dst[i] = src[i];
}

// ---------- voxel key + occupancy ----------
__global__ __launch_bounds__(256) void k_mark(const int* __restrict__ verts,
                                              u32* __restrict__ keys,
                                              u32* __restrict__ flag, int N) {
    int i = blockIdx.x * 256 + threadIdx.x;
    if (i >= N) return;
    u32 x = (u32)(verts[3*i+0] / POOLK);
    u32 y = (u32)(verts[3*i+1] / POOLK);
    u32 z = (u32)(verts[3*i+2] / POOLK);
    u32 k = x * (GDIM*GDIM) + y * GDIM + z;
    keys[i] = k;
    flag[k] = 1u;   // benign race: same value
}

// ---------- 3-kernel exclusive scan (1024 elems / block, <=256 blocks) ----------
__global__ __launch_bounds__(256) void k_scan_block(const u32* __restrict__ in,
                                                    u32* __restrict__ out,
                                                    u32* __restrict__ sums, int n) {
    __shared__ u32 sh[256];
    int t = threadIdx.x;
    int base = blockIdx.x * 1024 + t * 4;
    u32 a0=0,a1=0,a2=0,a3=0;
    if (base+0 < n) a0 = in[base+0];
    if (base+1 < n) a1 = in[base+1];
    if (base+2 < n) a2 = in[base+2];
    if (base+3 < n) a3 = in[base+3];
    u32 tsum = a0+a1+a2+a3;
    sh[t] = tsum; __syncthreads();
    for (int off = 1; off < 256; off <<= 1) {
        u32 v = sh[t];
        if (t >= off) v += sh[t-off];
        __syncthreads(); sh[t] = v; __syncthreads();
    }
    u32 incl = sh[t];
    u32 r = incl - tsum;              // exclusive
    if (t == 255) sums[blockIdx.x] = incl;
    if (base+0 < n) { out[base+0] = r; r += a0; }
    if (base+1 < n) { out[base+1] = r; r += a1; }
    if (base+2 < n) { out[base+2] = r; r += a2; }
    if (base+3 < n) { out[base+3] = r; }
}
__global__ __launch_bounds__(256) void k_scan_top(u32* sums, int nb, u32* total) {
    __shared__ u32 sh[256];
    int t = threadIdx.x;
    u32 x = (t < nb) ? sums[t] : 0u;
    sh[t] = x; __syncthreads();
    for (int off = 1; off < 256; off <<= 1) {
        u32 v = sh[t];
        if (t >= off) v += sh[t-off];
        __syncthreads(); sh[t] = v; __syncthreads();
    }
    if (t < nb) sums[t] = sh[t] - x;  // exclusive
    if (t == 255 && total) *total = sh[255];
}
__global__ __launch_bounds__(256) void k_scan_add(u32* out, const u32* sums, int n) {
    int i = blockIdx.x * 256 + threadIdx.x;
    if (i < n) out[i] += sums[i >> 10];
}

// ---------- inverse map ----------
__global__ __launch_bounds__(256) void k_inv(const u32* keys, const u32* rank,
                                             u32* inv, int N) {
    int i = blockIdx.x * 256 + threadIdx.x;
    if (i < N) inv[i] = rank[keys[i]];
}

// ---------- feature pooling: TDM stage -> LDS -> ordered-uint atomic max ----------
#if defined(__gfx1250__) && __has_builtin(__builtin_amdgcn_tensor_load_to_lds) && __has_builtin(__builtin_amdgcn_s_wait_tensorcnt)
#define USE_TDM 1
#else
#define USE_TDM 0
#endif

#if USE_TDM
typedef unsigned int u32x4 __attribute__((ext_vector_type(4)));
typedef int          i32x8 __attribute__((ext_vector_type(8)));
typedef int          i32x4 __attribute__((ext_vector_type(4)));
__device__ __forceinline__ u32 rfl(u32 v) {
    return (u32)__builtin_amdgcn_readfirstlane((int)v);
}

// 256 threads / block. Wave 0 issues ONE 16KB Tensor-Data-Mover DMA for the
// whole block tile (256 vertices x 16 feats, tile_dim0=4096 dwords), waits
// TENSORcnt, then the block barrier releases all 8 waves to consume LDS with
// a conflict-free (stride-1) mapping and scatter via global_atomic_max_u32.
// TDM ignores EXEC, so the issue is gated on a *scalar* wave-id branch to
// guarantee exactly one DMA per block.
__global__ __launch_bounds__(256) void k_pool_tdm(const float* __restrict__ feat,
                                                  const u32* __restrict__ inv,
                                                  u32* __restrict__ pooled, int N) {
    __shared__ float shm[4096];
    const long long vbase = (long long)blockIdx.x * 256;
    const long long ebase = vbase * 16;                 // dword base in features

    const int waveid = __builtin_amdgcn_readfirstlane((int)(threadIdx.x >> 5));
    if (waveid == 0) {                                  // scalar branch: one DMA
        const long long etot = (long long)N * 16;
        long long rem64 = etot - ebase;
        if (rem64 > 4096) rem64 = 4096;
        if (rem64 < 0)    rem64 = 0;
        const u32 remain = (u32)rem64;                  // OOB elems read as zero

        unsigned long long ga = (unsigned long long)(uintptr_t)(feat + ebase);
        const u32 ldsa = (u32)(uintptr_t)(&shm[0]);     // low 32b of flat shared = LDS byte addr

        // D# group0: count=1 | lds_addr | global_addr[56:0] | type=2
        u32x4 g0 = { rfl(1u), rfl(ldsa), rfl((u32)ga),
                     rfl(((u32)((ga >> 32) & 0x01FFFFFFull)) | 0x80000000u) };
        // D# group1: data_size=4B; tensor_dim0=remain (bits79:48); tile_dim0=4096 (bits127:112); 1-D
        i32x8 g1 = { (int)rfl(2u << 16),
                     (int)rfl((remain & 0xFFFFu) << 16),
                     (int)rfl((remain >> 16) & 0xFFFFu),
                     (int)rfl(4096u << 16),
                     0, 0, 0, 0 };
        i32x4 z4 = {0, 0, 0, 0};
        i32x8 z8 = {0, 0, 0, 0, 0, 0, 0, 0};
        __builtin_amdgcn_tensor_load_to_lds(g0, g1, z4, z4, z8, 0);
        __builtin_amdgcn_s_wait_tensorcnt(0);
    }
    __syncthreads();
    asm volatile("" ::: "memory");                      // TDM wrote LDS behind compiler's back

    #pragma unroll
    for (int j = 0; j < 16; ++j) {
        const int idx = j * 256 + (int)threadIdx.x;     // stride-1 across lanes: no LDS bank conflicts
        const long long v = vbase + (idx >> 4);
        if (v < N) {
            const int f = idx & 15;
            atomicMax(&pooled[(long long)inv[v] * 16 + f], f2ord(shm[idx]));
        }
    }
}
#endif

__global__ __launch_bounds__(256) void k_pool_gen(const float* __restrict__ feat,
                                                  const u32* __restrict__ inv,
                                                  u32* __restrict__ pooled,
                                                  long long NF, int F) {
    long long idx = (long long)blockIdx.x * 256 + threadIdx.x;
    if (idx >= NF) return;
    long long i = idx / F; int f = (int)(idx - i * F);
    atomicMax(&pooled[(long long)inv[i] * F + f], f2ord(feat[idx]));
}

// ---------- outputs: positions, self-loops, pooled ----------
__global__ __launch_bounds__(256) void k_scatter_vox(const u32* __restrict__ flag,
                                                     const u32* __restrict__ rank,
                                                     float* __restrict__ pos,
                                                     float* __restrict__ eout,
                                                     long long E) {
    int k = blockIdx.x * 256 + threadIdx.x;
    if (k >= G3 || !flag[k]) return;
    u32 r = rank[k];
    pos[(long long)r*3 + 0] = (float)(k / (GDIM*GDIM));
    pos[(long long)r*3 + 1] = (float)((k / GDIM) % GDIM);
    pos[(long long)r*3 + 2] = (float)(k % GDIM);
    long long row = E + r;
    eout[row*2 + 0] = (float)r;
    eout[row*2 + 1] = (float)r;
}
__global__ __launch_bounds__(256) void k_pooled_out(const u32* __restrict__ pooled,
                                                    const u32* __restrict__ scal,
                                                    float* __restrict__ out, int F) {
    long long idx = (long long)blockIdx.x * 256 + threadIdx.x;
    long long tot = (long long)G3 * F;
    if (idx >= tot) return;
    long long j = idx / F;
    out[idx] = (j < (long long)scal[0]) ? ord2f(pooled[idx]) : 0.0f;
}

// ---------- edges: count -> scatter -> per-bucket sort+dedup -> emit ----------
__global__ __launch_bounds__(256) void k_edge_count(const int* __restrict__ edges,
                                                    const u32* __restrict__ inv,
                                                    u32* __restrict__ bcnt, long long E) {
    long long i = (long long)blockIdx.x * 256 + threadIdx.x;
    if (i >= E) return;
    u32 a = inv[edges[2*i]], b = inv[edges[2*i+1]];
    if (a != b) atomicAdd(&bcnt[a], 1u);
}
__global__ __launch_bounds__(256) void k_edge_scatter(const int* __restrict__ edges,
                                                      const u32* __restrict__ inv,
                                                      u32* __restrict__ bcur,
                                                      u32* __restrict__ sb, long long E) {
    long long i = (long long)blockIdx.x * 256 + threadIdx.x;
    if (i >= E) return;
    u32 a = inv[edges[2*i]], b = inv[edges[2*i+1]];
    if (a != b) sb[atomicAdd(&bcur[a], 1u)] = b;
}
__global__ __launch_bounds__(256) void k_sort_dedup(const u32* __restrict__ bcnt,
                                                    const u32* __restrict__ boff,
                                                    u32* __restrict__ sb,
                                                    u32* __restrict__ ucnt) {
    int a = blockIdx.x * 256 + threadIdx.x;
    if (a >= G3) return;
    int len = (int)bcnt[a];
    if (len == 0) { ucnt[a] = 0u; return; }
    u32 s = boff[a];
    for (int i = 1; i < len; ++i) {               // insertion sort (avg len ~31)
        u32 key = sb[s+i]; int j = i - 1;
        while (j >= 0 && sb[s+j] > key) { sb[s+j+1] = sb[s+j]; --j; }
        sb[s+j+1] = key;
    }
    int m = 1;
    for (int i = 1; i < len; ++i)
        if (sb[s+i] != sb[s+m-1]) sb[s + m++] = sb[s+i];
    ucnt[a] = (u32)m;
}
__global__ __launch_bounds__(256) void k_edge_emit(const u32* __restrict__ ucnt,
                                                   const u32* __restrict__ uoff,
                                                   const u32* __restrict__ boff,
                                                   const u32* __restrict__ sb,
                                                   float* __restrict__ eout) {
    int a = blockIdx.x * 256 + threadIdx.x;
    if (a >= G3) return;
    int m = (int)ucnt[a];
    if (!m) return;
    u32 o = uoff[a], s = boff[a];
    for (int k = 0; k < m; ++k) {
        long long row = (long long)o + k;
        eout[row*2 + 0] = (float)a;
        eout[row*2 + 1] = (float)sb[s + k];
    }
}
__global__ void k_scalars(const u32* scal, float* out2) {
    if (threadIdx.x == 0 && blockIdx.x == 0) {
        u32 nu = scal[0], nd = scal[1];
        out2[0] = (float)nu;
        out2[1] = (float)(nd + nu);
    }
}

// ---------------------------------------------------------------------------
extern "C" void kernel_launch(void* const* d_in, const int* in_sizes, int n_in,
                              void* d_out, int out_size, void* d_ws, size_t ws_size,
                              hipStream_t stream) {
    const int  N = in_sizes[0] / 3;
    const int  F = (N > 0) ? in_sizes[1] / N : 16;
    const long long E = in_sizes[2] / 2;

    const int*   verts = (const int*)d_in[0];
    const float* feat  = (const float*)d_in[1];
    const int*   edges = (const int*)d_in[2];
    float* out = (float*)d_out;

    // workspace layout (u32 units); zero-needed arrays first (contiguous)
    u32* W      = (u32*)d_ws;
    u32* flag   = W;                         // G3
    u32* bcnt   = flag + G3;                 // G3
    u32* ucnt   = bcnt + G3;                 // G3
    u32* pooled = ucnt + G3;                 // G3*F
    u32* rank   = pooled + (long long)G3*F;  // G3
    u32* boff   = rank + G3;                 // G3
    u32* bcur   = boff + G3;                 // G3
    u32* uoff   = bcur + G3;                 // G3
    u32* sums   = uoff + G3;                 // 256
    u32* scal   = sums + 256;                // 16  [0]=num_unique [1]=num_dedup [2]=scratch
    u32* keys   = scal + 16;                 // N
    u32* inv    = keys + N;                  // N
    u32* sb     = inv + N;                   // E

    float* out_pos  = out;
    float* out_pool = out_pos  + (long long)G3 * 3;
    float* out_edge = out_pool + (long long)G3 * F;
    float* out_scal = out_edge + (E + (long long)G3) * 2;

    const int nbG3 = (G3 + 1023) / 1024;     // 256

    // 1) zero occupancy / histograms / pooled accumulator
    k_zero_u32<<<2048, 256, 0, stream>>>(W, (long long)G3 * (3 + F));
    // 2) voxel keys + occupancy bitmap
    k_mark<<<(N + 255)/256, 256, 0, stream>>>(verts, keys, flag, N);
    // 3) exclusive scan flag -> rank, total -> num_unique
    k_scan_block<<<nbG3, 256, 0, stream>>>(flag, rank, sums, G3);
    k_scan_top  <<<1,    256, 0, stream>>>(sums, nbG3, scal + 0);
    k_scan_add  <<<(G3 + 255)/256, 256, 0, stream>>>(rank, sums, G3);
    // 4) inverse map per vertex
    k_inv<<<(N + 255)/256, 256, 0, stream>>>(keys, rank, inv, N);
    // 5) segment-max pooling (TDM-staged when F==16)
#if USE_TDM
    if (F == 16)
        k_pool_tdm<<<(N + 255)/256, 256, 0, stream>>>(feat, inv, pooled, N);
    else
#endif
        k_pool_gen<<<(int)(((long long)N*F + 255)/256), 256, 0, stream>>>(feat, inv, pooled, (long long)N*F, F);
    // 6) output defaults (padding rows)
    k_fill_f32<<<2048, 256, 0, stream>>>(out_pos, 0.0f, (long long)G3 * 3);
    k_fill_f32<<<2048, 256, 0, stream>>>(out_edge, -1.0f, (E + (long long)G3) * 2);
    // 7) positions + self loops for occupied voxels
    k_scatter_vox<<<(G3 + 255)/256, 256, 0, stream>>>(flag, rank, out_pos, out_edge, E);
    // 8) pooled output (decode ordered uints; empty rows -> 0)
    k_pooled_out<<<(int)(((long long)G3*F + 255)/256), 256, 0, stream>>>(pooled, scal, out_pool, F);
    // 9) edge counting sort by source voxel id
    k_edge_count<<<(int)((E + 255)/256), 256, 0, stream>>>(edges, inv, bcnt, E);
    k_scan_block<<<nbG3, 256, 0, stream>>>(bcnt, boff, sums, G3);
    k_scan_top  <<<1,    256, 0, stream>>>(sums, nbG3, scal + 2);
    k_scan_add  <<<(G3 + 255)/256, 256, 0, stream>>>(boff, sums, G3);
    k_copy_u32  <<<(G3 + 255)/256, 256, 0, stream>>>(bcur, boff, G3);
    k_edge_scatter<<<(int)((E + 255)/256), 256, 0, stream>>>(edges, inv, bcur, sb, E);
    // 10) per-bucket sort + dedup, then compact emission in sorted (a,b) order
    k_sort_dedup<<<(G3 + 255)/256, 256, 0, stream>>>(bcnt, boff, sb, ucnt);
    k_scan_block<<<nbG3, 256, 0, stream>>>(ucnt, uoff, sums, G3);
    k_scan_top  <<<1,    256, 0, stream>>>(sums, nbG3, scal + 1);
    k_scan_add  <<<(G3 + 255)/256, 256, 0, stream>>>(uoff, sums, G3);
    k_edge_emit<<<(G3 + 255)/256, 256, 0, stream>>>(ucnt, uoff, boff, sb, out_edge);
    // 11) counts
    k_scalars<<<1, 32, 0, stream>>>(scal, out_scal);
}